// Spherical_Voxelization_16733192585422
// MI455X (gfx1250) — compile-verified
//
#include <hip/hip_runtime.h>
#include <math.h>

// Problem constants (from reference): B=8, C=64, N=65536, R=32
#define B_   8
#define C_   64
#define N_   65536
#define R_   32
#define R3_  32768
#define EPSF 1e-20f
#define PI_F 3.14159265358979323846f

// ---------------------------------------------------------------------------
// CDNA5 async global->LDS path (gfx1250). b32 builtin takes int* args,
// b128 takes int __attribute__((vector_size(16)))* args; clang inserts the
// generic->AS1/AS3 addrspacecasts itself.
// ---------------------------------------------------------------------------
typedef int v4i __attribute__((vector_size(16)));

#if defined(__HIP_DEVICE_COMPILE__)
#  if __has_builtin(__builtin_amdgcn_global_load_async_to_lds_b128)
#    define HAVE_ASYNC_B128 1
#  endif
#  if __has_builtin(__builtin_amdgcn_global_load_async_to_lds_b32)
#    define HAVE_ASYNC_B32 1
#  endif
#  if __has_builtin(__builtin_amdgcn_s_wait_asynccnt)
#    define WAITA(n) __builtin_amdgcn_s_wait_asynccnt(n)
#  else
#    define WAITA(n) asm volatile("s_wait_asynccnt %0" :: "n"(n) : "memory")
#  endif
#endif

__device__ __forceinline__ int clamp_r(int v) {
  return v < 0 ? 0 : (v > (R_ - 1) ? (R_ - 1) : v);
}

// ---------------------------------------------------------------------------
// K0: zero the per-(batch,voxel) counts in workspace
// ---------------------------------------------------------------------------
extern "C" __global__ void __launch_bounds__(256)
k_zero_cnt(unsigned int* __restrict__ cnt) {
  cnt[(size_t)blockIdx.x * 256 + threadIdx.x] = 0u;
}

// ---------------------------------------------------------------------------
// K1: per-batch mean (x,y,z) and max ||c - mean|| (one workgroup per batch)
// ---------------------------------------------------------------------------
extern "C" __global__ void __launch_bounds__(1024)
k_stats(const float* __restrict__ coords, float* __restrict__ mean,
        float* __restrict__ scale) {
  const int b = blockIdx.x;
  const int t = threadIdx.x;
  const float* cb = coords + (size_t)b * 3 * N_;

  __shared__ float sx[1024], sy[1024], sz[1024];
  __shared__ float mxyz[3];

  float ax = 0.f, ay = 0.f, az = 0.f;
  for (int n = t; n < N_; n += 1024) {
    ax += cb[n];
    ay += cb[N_ + n];
    az += cb[2 * N_ + n];
  }
  sx[t] = ax; sy[t] = ay; sz[t] = az;
  __syncthreads();
  for (int s = 512; s > 0; s >>= 1) {
    if (t < s) { sx[t] += sx[t + s]; sy[t] += sy[t + s]; sz[t] += sz[t + s]; }
    __syncthreads();
  }
  if (t == 0) {
    mxyz[0] = sx[0] * (1.0f / N_);
    mxyz[1] = sy[0] * (1.0f / N_);
    mxyz[2] = sz[0] * (1.0f / N_);
  }
  __syncthreads();
  const float mx = mxyz[0], my = mxyz[1], mz = mxyz[2];

  float mm = 0.f;
  for (int n = t; n < N_; n += 1024) {
    const float x = cb[n] - mx;
    const float y = cb[N_ + n] - my;
    const float z = cb[2 * N_ + n] - mz;
    mm = fmaxf(mm, x * x + y * y + z * z);
  }
  sx[t] = mm;
  __syncthreads();
  for (int s = 512; s > 0; s >>= 1) {
    if (t < s) sx[t] = fmaxf(sx[t], sx[t + s]);
    __syncthreads();
  }
  if (t == 0) {
    mean[b * 4 + 0] = mx;
    mean[b * 4 + 1] = my;
    mean[b * 4 + 2] = mz;
    scale[b] = sqrtf(sx[0]);
  }
}

// ---------------------------------------------------------------------------
// K2: normalized coords, spherical voxel index, count histogram (global u32)
// ---------------------------------------------------------------------------
extern "C" __global__ void __launch_bounds__(256)
k_index(const float* __restrict__ coords, const float* __restrict__ mean,
        const float* __restrict__ scale, float* __restrict__ nco,
        int* __restrict__ inds, int* __restrict__ idxws,
        unsigned int* __restrict__ cnt) {
  const int gid = blockIdx.x * 256 + threadIdx.x;  // [0, B*N)
  const int b = gid >> 16;                         // N == 2^16
  const int n = gid & (N_ - 1);
  const float* cb = coords + (size_t)b * 3 * N_;

  const float inv = 1.0f / (scale[b] + EPSF);
  const float mx = mean[b * 4 + 0];
  const float my = mean[b * 4 + 1];
  const float mz = mean[b * 4 + 2];

  const float x = (cb[n] - mx) * inv;
  const float y = (cb[N_ + n] - my) * inv;
  const float z = (cb[2 * N_ + n] - mz) * inv;

  float* ncb = nco + (size_t)b * 3 * N_;
  ncb[n]          = x;
  ncb[N_ + n]     = y;
  ncb[2 * N_ + n] = z;

  const float rho = sqrtf(x * x + y * y + z * z);
  const int ri = clamp_r((int)floorf(rho * (float)R_));
  const float ct = fminf(fmaxf(z / (rho + EPSF), -1.0f), 1.0f);
  const int ti = clamp_r((int)floorf(acosf(ct) * ((float)R_ / PI_F)));
  const float phi = atan2f(y, x) + PI_F;
  const int pi_ = clamp_r((int)floorf(phi * ((float)R_ / (2.0f * PI_F))));

  const int idx = (ri * R_ + ti) * R_ + pi_;
  inds[gid]  = idx;  // int32 bits into the flat output region
  idxws[gid] = idx;
  atomicAdd(&cnt[(size_t)b * R3_ + idx], 1u);
}

// ---------------------------------------------------------------------------
// K3: reciprocal counts  cntinv = 1 / max(cnt, 1)
// ---------------------------------------------------------------------------
extern "C" __global__ void __launch_bounds__(256)
k_cntinv(const unsigned int* __restrict__ cnt, float* __restrict__ cntinv) {
  const size_t i = (size_t)blockIdx.x * 256 + threadIdx.x;
  const unsigned int c = cnt[i];
  cntinv[i] = 1.0f / (float)(c > 1u ? c : 1u);
}

// ---------------------------------------------------------------------------
// K4: one workgroup per (batch, channel). Full 32768-bin f32 histogram in
// LDS (128 KB, dynamic). Feature/index tiles (1024 points/chunk) double-
// buffered via async global->LDS b128 loads (16 B/lane/op, ASYNCcnt
// pipelined), consumed as ds_load_b128 + 4x ds_add_f32, scaled coalesced
// writeout. LDS total 144 KB -> two workgroups per 320 KB WGP.
// ---------------------------------------------------------------------------
extern "C" __global__ void __launch_bounds__(256)
k_accum(const float* __restrict__ features, const int* __restrict__ idxws,
        const float* __restrict__ cntinv, float* __restrict__ out) {
  extern __shared__ float fsum[];                     // R3_ floats = 128 KB
  __shared__ __align__(16) float sFeat[2][1024];      // 8 KB
  __shared__ __align__(16) int   sIdx[2][1024];       // 8 KB

  const int t  = threadIdx.x;
  const int bc = blockIdx.x;           // b*C + c
  const int b  = bc >> 6;              // C == 64
  const float* featp = features + (size_t)bc * N_;
  const int*   idxp  = idxws + (size_t)b * N_;

  for (int v = t; v < R3_; v += 256) fsum[v] = 0.0f;
  __syncthreads();

  // Each thread owns 4 consecutive slots [4t, 4t+3] of the 1024-wide tile.
  auto stage = [&](int ch, int bf) {
    const int n_ = (t << 2) + (ch << 10);
#if defined(HAVE_ASYNC_B128)
    __builtin_amdgcn_global_load_async_to_lds_b128(
        (v4i*)(featp + n_), (v4i*)&sFeat[bf][t << 2], 0, 0);
    __builtin_amdgcn_global_load_async_to_lds_b128(
        (v4i*)(idxp + n_), (v4i*)&sIdx[bf][t << 2], 0, 0);
#elif defined(HAVE_ASYNC_B32)
    for (int j = 0; j < 4; ++j) {
      __builtin_amdgcn_global_load_async_to_lds_b32(
          (int*)(featp + n_ + j), (int*)&sFeat[bf][(t << 2) + j], 0, 0);
      __builtin_amdgcn_global_load_async_to_lds_b32(
          (int*)(idxp + n_ + j), (int*)&sIdx[bf][(t << 2) + j], 0, 0);
    }
#else
    __builtin_prefetch(featp + n_ + 4096, 0, 1);  // global_prefetch_b8
    *(float4*)&sFeat[bf][t << 2] = *(const float4*)(featp + n_);
    *(int4*)&sIdx[bf][t << 2]    = *(const int4*)(idxp + n_);
#endif
  };

  const int kChunks = N_ / 1024;       // 64 chunks of 1024 points
  stage(0, 0);
  for (int k = 0; k < kChunks; ++k) {
    const int bf = k & 1;
    if (k + 1 < kChunks) {
      stage(k + 1, bf ^ 1);            // next chunk's async ops in flight
#if defined(HAVE_ASYNC_B128)
      WAITA(2);                        // chunk k's 2 ops done (in-order)
#elif defined(HAVE_ASYNC_B32)
      WAITA(8);
#endif
    } else {
#if defined(HAVE_ASYNC_B128) || defined(HAVE_ASYNC_B32)
      WAITA(0);
#endif
    }
    asm volatile("" ::: "memory");     // keep LDS reads after the wait
    const int4   iv = *(const int4*)&sIdx[bf][t << 2];    // ds_load_b128
    const float4 fv = *(const float4*)&sFeat[bf][t << 2]; // ds_load_b128
    atomicAdd(&fsum[iv.x], fv.x);      // ds_add_f32 (no return)
    atomicAdd(&fsum[iv.y], fv.y);
    atomicAdd(&fsum[iv.z], fv.z);
    atomicAdd(&fsum[iv.w], fv.w);
  }
  __syncthreads();

  const float* cinv = cntinv + (size_t)b * R3_;
  float* op = out + (size_t)bc * R3_;
  for (int v = t; v < R3_; v += 256) op[v] = fsum[v] * cinv[v];
}

// ---------------------------------------------------------------------------
// Host launcher
// ---------------------------------------------------------------------------
extern "C" void kernel_launch(void* const* d_in, const int* in_sizes, int n_in,
                              void* d_out, int out_size, void* d_ws, size_t ws_size,
                              hipStream_t stream) {
  (void)in_sizes; (void)n_in; (void)out_size; (void)ws_size;

  const float* features = (const float*)d_in[0];  // (B, C, N) f32
  const float* coords   = (const float*)d_in[1];  // (B, 3, N) f32

  float* out  = (float*)d_out;                              // (B,C,32,32,32)
  int*   inds = (int*)(out + (size_t)B_ * C_ * R3_);        // (B,N) int32 bits
  float* nco  = out + (size_t)B_ * C_ * R3_ + (size_t)B_ * N_;  // (B,3,N)

  // Workspace layout (~4.2 MB)
  float*        mean   = (float*)d_ws;                       // B*4
  float*        scale  = mean + B_ * 4;                      // B
  unsigned int* cnt    = (unsigned int*)(scale + B_);        // B*R3
  float*        cntinv = (float*)(cnt + (size_t)B_ * R3_);   // B*R3
  int*          idxws  = (int*)(cntinv + (size_t)B_ * R3_);  // B*N

  k_zero_cnt<<<dim3((B_ * R3_) / 256), dim3(256), 0, stream>>>(cnt);
  k_stats<<<dim3(B_), dim3(1024), 0, stream>>>(coords, mean, scale);
  k_index<<<dim3((B_ * N_) / 256), dim3(256), 0, stream>>>(
      coords, mean, scale, nco, inds, idxws, cnt);
  k_cntinv<<<dim3((B_ * R3_) / 256), dim3(256), 0, stream>>>(cnt, cntinv);
  k_accum<<<dim3(B_ * C_), dim3(256), R3_ * sizeof(float), stream>>>(
      features, idxws, cntinv, out);
}